// RPN_56667798503871
// MI455X (gfx1250) — compile-verified
//
#include <hip/hip_runtime.h>
#include <hip/hip_bf16.h>
#include <stdint.h>

// ---------------- problem constants (match reference) ----------------
#define M_GT            40
#define FG_TH           0.7f
#define BG_TH           0.3f
#define BETA_C          (1.0f/9.0f)
#define BATCH_PER_IMG   256
#define POS_TARGET      128      // BATCH_PER_IMAGE * POS_FRACTION
#define BLOCK           256
#define ITERS           4
#define CHUNK           (BLOCK*ITERS)   // 1024 anchors per block -> 245 blocks/batch (<=256 for k_scan)

// ---------------- CDNA5 TDM (tensor DMA) guard ----------------
#if defined(__HIP_DEVICE_COMPILE__) && \
    __has_builtin(__builtin_amdgcn_tensor_load_to_lds) && \
    __has_builtin(__builtin_amdgcn_s_wait_tensorcnt)
#define USE_TDM 1
#else
#define USE_TDM 0
#endif

typedef unsigned int u32x4 __attribute__((ext_vector_type(4)));
typedef int          i32x4 __attribute__((ext_vector_type(4)));
typedef int          i32x8 __attribute__((ext_vector_type(8)));

// DMA a contiguous run of `nelem` f32 from global to LDS via the Tensor Data
// Mover (1-D tensor, 1-D tile, data_size=4B). Falls back to plain loads when
// the builtin is unavailable. Caller must __syncthreads() afterwards.
__device__ __forceinline__ void tdm_or_copy_to_lds(const float* gsrc, float* ldst, int nelem) {
#if USE_TDM
  if (threadIdx.x < 32) {   // one wave issues the DMA (EXEC full for wave 0)
    unsigned long long ga = (unsigned long long)(uintptr_t)gsrc;
    unsigned lds_off = (unsigned)(uintptr_t)ldst;  // low 32 bits of LDS-aperture flat addr = LDS byte offset
    u32x4 g0;
    g0[0] = 1u;                                            // count=1 (valid user D#)
    g0[1] = lds_off;                                       // lds_addr
    g0[2] = (unsigned)(ga & 0xFFFFFFFFull);                // global_addr[31:0]
    g0[3] = (unsigned)((ga >> 32) & 0x01FFFFFFull)         // global_addr[56:32]
            | (2u << 30);                                  // type=2 ("image")
    i32x8 g1;
    g1[0] = (int)(2u << 16);                               // data_size=2 (4 bytes)
    g1[1] = (int)(((unsigned)nelem & 0xFFFFu) << 16);      // tensor_dim0[15:0] @ bits 63:48
    g1[2] = (int)(((unsigned)nelem >> 16) | (1u << 16));   // tensor_dim0[31:16] | tensor_dim1=1
    g1[3] = (int)(((unsigned)nelem & 0xFFFFu) << 16);      // tile_dim0 @ bits 127:112
    g1[4] = 0;                                             // tile_dim1=0, tile_dim2=0 (unused)
    g1[5] = nelem;                                         // tensor_dim0_stride[31:0]
    g1[6] = 0;
    g1[7] = 0;
    i32x4 gz = {0, 0, 0, 0};
#if __has_include(<hip/amd_detail/amd_gfx1250_TDM.h>)
    i32x8 gz8 = {0, 0, 0, 0, 0, 0, 0, 0};
    __builtin_amdgcn_tensor_load_to_lds(g0, g1, gz, gz, gz8, 0);   // 6-arg toolchain
#else
    __builtin_amdgcn_tensor_load_to_lds(g0, g1, gz, gz, 0);        // 5-arg (ROCm 7.2)
#endif
    __builtin_amdgcn_s_wait_tensorcnt(0);
  }
#else
  for (int k = (int)threadIdx.x; k < nelem; k += (int)blockDim.x) ldst[k] = gsrc[k];
#endif
}

// ---------------- shared FP helpers (bit-identical across kernels) ----------------
__device__ __forceinline__ float box_area(float x1, float y1, float x2, float y2) {
#pragma clang fp contract(off)
  return (x2 - x1 + 1.0f) * (y2 - y1 + 1.0f);
}

__device__ __forceinline__ float iou_one(float ax1, float ay1, float ax2, float ay2, float area_a,
                                         float gx1, float gy1, float gx2, float gy2, float area_g) {
#pragma clang fp contract(off)
  float lx = fmaxf(gx1, ax1);
  float ly = fmaxf(gy1, ay1);
  float rx = fminf(gx2, ax2);
  float ry = fminf(gy2, ay2);
  float w  = fmaxf(rx - lx + 1.0f, 0.0f);
  float h  = fmaxf(ry - ly + 1.0f, 0.0f);
  float inter = w * h;
  return inter / (area_g + area_a - inter);
}

// label: 1=pos, 0=neg, -1=ignore ; argm = argmax_m iou (first max, as jnp.argmax)
__device__ __forceinline__ void anchor_match(float4 a, const float* s_gt, const float* s_ga,
                                             const float* s_hi, int vis, int& label, int& argm) {
#pragma clang fp contract(off)
  float area_a = box_area(a.x, a.y, a.z, a.w);
  float best = -1.0f; int bi = 0; bool isb = false;
#pragma unroll 8
  for (int m = 0; m < M_GT; ++m) {
    float v = iou_one(a.x, a.y, a.z, a.w, area_a,
                      s_gt[4*m+0], s_gt[4*m+1], s_gt[4*m+2], s_gt[4*m+3], s_ga[m]);
    if (v > best) { best = v; bi = m; }
    isb = isb || (v == s_hi[m]);   // allow_low_quality_matches
  }
  int matches = (best < BG_TH) ? -1 : ((best < FG_TH) ? -2 : bi);
  if (isb) matches = bi;
  label = (matches == -2) ? -1 : ((vis == 0) ? -1 : ((matches >= 0) ? 1 : 0));
  argm = bi;
}

__device__ __forceinline__ float wave_sumf(float v) {
  for (int o = 16; o; o >>= 1) v += __shfl_xor(v, o, 32);
  return v;
}
__device__ __forceinline__ int wave_sumi(int v) {
  for (int o = 16; o; o >>= 1) v += __shfl_xor(v, o, 32);
  return v;
}

// ---------------- kernels (TDM user first so disasm snippet shows tensor_load_to_lds) ----------------

// highest_per_gt[b][m] = max_n iou  (uint atomicMax on non-negative float bits: exact)
__global__ void k_gtmax(const float4* __restrict__ anc, const float* __restrict__ gt,
                        unsigned* __restrict__ highest, int N) {
  __shared__ __align__(16) float s_gt[M_GT * 4];
  __shared__ float s_ga[M_GT];
  __shared__ float s_red[M_GT * 8];
  const int b = blockIdx.y;
  tdm_or_copy_to_lds(gt + b * M_GT * 4, s_gt, M_GT * 4);
  __syncthreads();
  if (threadIdx.x < M_GT) {
    int m = threadIdx.x;
    s_ga[m] = box_area(s_gt[4*m+0], s_gt[4*m+1], s_gt[4*m+2], s_gt[4*m+3]);
  }
  __syncthreads();

  float mx[M_GT];
#pragma unroll
  for (int m = 0; m < M_GT; ++m) mx[m] = 0.0f;

  const int bN = b * N;
  const int base = blockIdx.x * CHUNK;
  for (int it = 0; it < ITERS; ++it) {
    int i = base + it * BLOCK + (int)threadIdx.x;
    if (i < N) {
      float4 a = anc[bN + i];
      if (it + 1 < ITERS) __builtin_prefetch(&anc[bN + i + BLOCK], 0, 0);
      float area_a = box_area(a.x, a.y, a.z, a.w);
#pragma unroll
      for (int m = 0; m < M_GT; ++m) {
        float v = iou_one(a.x, a.y, a.z, a.w, area_a,
                          s_gt[4*m+0], s_gt[4*m+1], s_gt[4*m+2], s_gt[4*m+3], s_ga[m]);
        mx[m] = fmaxf(mx[m], v);
      }
    }
  }
  const int lane = threadIdx.x & 31, wv = threadIdx.x >> 5;
#pragma unroll
  for (int m = 0; m < M_GT; ++m) {
    float v = mx[m];
    for (int o = 16; o; o >>= 1) v = fmaxf(v, __shfl_xor(v, o, 32));
    if (lane == 0) s_red[m * 8 + wv] = v;
  }
  __syncthreads();
  if (threadIdx.x < M_GT) {          // one atomic per block per gt (40/block total)
    int m = threadIdx.x;
    float v = s_red[m * 8 + 0];
    for (int w = 1; w < 8; ++w) v = fmaxf(v, s_red[m * 8 + w]);
    atomicMax(&highest[b * M_GT + m], __float_as_uint(v));
  }
}

__global__ void k_init(unsigned* __restrict__ highest, int n) {
  int t = blockIdx.x * blockDim.x + threadIdx.x;
  if (t < n) highest[t] = 0u;   // 0.0f bits; IoU >= 0 so this is the identity for max
}

// per-block pos/neg counts
__global__ void k_count(const float4* __restrict__ anc, const float* __restrict__ gt,
                        const float* __restrict__ highestF, const int* __restrict__ vis,
                        int* __restrict__ blockPos, int* __restrict__ blockNeg,
                        int N, int nblk) {
  __shared__ __align__(16) float s_gt[M_GT * 4];
  __shared__ float s_ga[M_GT];
  __shared__ float s_hi[M_GT];
  __shared__ int s_wp[8], s_wn[8];
  const int b = blockIdx.y;
  tdm_or_copy_to_lds(gt + b * M_GT * 4, s_gt, M_GT * 4);
  if (threadIdx.x < M_GT) s_hi[threadIdx.x] = highestF[b * M_GT + threadIdx.x];
  __syncthreads();
  if (threadIdx.x < M_GT) {
    int m = threadIdx.x;
    s_ga[m] = box_area(s_gt[4*m+0], s_gt[4*m+1], s_gt[4*m+2], s_gt[4*m+3]);
  }
  __syncthreads();

  int cp = 0, cn = 0;
  const int bN = b * N;
  const int base = blockIdx.x * CHUNK;
  for (int it = 0; it < ITERS; ++it) {
    int i = base + it * BLOCK + (int)threadIdx.x;
    if (i < N) {
      float4 a = anc[bN + i];
      if (it + 1 < ITERS) __builtin_prefetch(&anc[bN + i + BLOCK], 0, 0);
      int label, arg;
      anchor_match(a, s_gt, s_ga, s_hi, vis[bN + i], label, arg);
      cp += (label == 1);
      cn += (label == 0);
    }
  }
  cp = wave_sumi(cp);
  cn = wave_sumi(cn);
  const int lane = threadIdx.x & 31, wv = threadIdx.x >> 5;
  if (lane == 0) { s_wp[wv] = cp; s_wn[wv] = cn; }
  __syncthreads();
  if (threadIdx.x == 0) {
    int sp = 0, sn = 0;
    for (int w = 0; w < 8; ++w) { sp += s_wp[w]; sn += s_wn[w]; }
    blockPos[b * nblk + blockIdx.x] = sp;
    blockNeg[b * nblk + blockIdx.x] = sn;
  }
}

// parallel exclusive scan over block counts (nblk <= 256) + num_pos/num_neg per batch
__global__ void k_scan(const int* __restrict__ blockPos, const int* __restrict__ blockNeg,
                       int* __restrict__ posOff, int* __restrict__ negOff,
                       int* __restrict__ numPN, int nblk) {
  __shared__ int sp[BLOCK], sn[BLOCK];
  const int b = blockIdx.x;
  const int t = threadIdx.x;
  const int vp = (t < nblk) ? blockPos[b * nblk + t] : 0;
  const int vn = (t < nblk) ? blockNeg[b * nblk + t] : 0;
  sp[t] = vp; sn[t] = vn;
  __syncthreads();
  for (int off = 1; off < BLOCK; off <<= 1) {      // Hillis-Steele inclusive scan
    int ap = (t >= off) ? sp[t - off] : 0;
    int an = (t >= off) ? sn[t - off] : 0;
    __syncthreads();
    sp[t] += ap; sn[t] += an;
    __syncthreads();
  }
  if (t < nblk) {
    posOff[b * nblk + t] = sp[t] - vp;             // exclusive
    negOff[b * nblk + t] = sn[t] - vn;
  }
  if (t == 0) {
    int pt = sp[BLOCK - 1], nt = sn[BLOCK - 1];    // totals (tail entries are zero)
    int np = pt < POS_TARGET ? pt : POS_TARGET;
    int rem = BATCH_PER_IMG - np;
    int nn = nt < rem ? nt : rem;
    numPN[b * 2 + 0] = np;
    numPN[b * 2 + 1] = nn;
  }
}

// first-k sampling via exact global ranks (ballot scan) + loss accumulation
__global__ void k_loss(const float* __restrict__ obj, const float4* __restrict__ breg,
                       const float4* __restrict__ anc, const float* __restrict__ gt,
                       const float* __restrict__ highestF, const int* __restrict__ vis,
                       const int* __restrict__ posOff, const int* __restrict__ negOff,
                       const int* __restrict__ numPN, float* __restrict__ partials,
                       int N, int nblk) {
  __shared__ __align__(16) float s_gt[M_GT * 4];
  __shared__ float s_ga[M_GT];
  __shared__ float s_hi[M_GT];
  __shared__ int s_wp[8], s_wn[8];
  __shared__ float s_ro[8], s_rb[8];
  const int b = blockIdx.y, bx = blockIdx.x;
  tdm_or_copy_to_lds(gt + b * M_GT * 4, s_gt, M_GT * 4);
  if (threadIdx.x < M_GT) s_hi[threadIdx.x] = highestF[b * M_GT + threadIdx.x];
  __syncthreads();
  if (threadIdx.x < M_GT) {
    int m = threadIdx.x;
    s_ga[m] = box_area(s_gt[4*m+0], s_gt[4*m+1], s_gt[4*m+2], s_gt[4*m+3]);
  }
  __syncthreads();

  const int np = numPN[b*2+0], nn = numPN[b*2+1];
  int posBase = posOff[b*nblk+bx], negBase = negOff[b*nblk+bx];
  const int bN = b * N;
  const int base = bx * CHUNK;
  const int lane = threadIdx.x & 31, wv = threadIdx.x >> 5;
  const unsigned ltmask = (1u << lane) - 1u;
  float obj_acc = 0.0f, box_acc = 0.0f;

  for (int it = 0; it < ITERS; ++it) {
    const int i = base + it * BLOCK + (int)threadIdx.x;
    const bool inR = i < N;
    int label = -1, arg = 0;
    float4 a = {0.0f, 0.0f, 0.0f, 0.0f};
    if (inR) {
      a = anc[bN + i];
      if (it + 1 < ITERS) __builtin_prefetch(&anc[bN + i + BLOCK], 0, 0);
      anchor_match(a, s_gt, s_ga, s_hi, vis[bN + i], label, arg);
    }
    const bool pos = inR && (label == 1);
    const bool neg = inR && (label == 0);
    const unsigned mp = (unsigned)__ballot(pos);
    const unsigned mn = (unsigned)__ballot(neg);
    if (lane == 0) { s_wp[wv] = __popc(mp); s_wn[wv] = __popc(mn); }
    __syncthreads();
    int offP = 0, offN = 0, totP = 0, totN = 0;
#pragma unroll
    for (int w = 0; w < 8; ++w) {
      int cpw = s_wp[w], cnw = s_wn[w];
      if (w < wv) { offP += cpw; offN += cnw; }
      totP += cpw; totN += cnw;
    }
    __syncthreads();   // protect s_wp/s_wn before next iteration's writes
    const int rankP = posBase + offP + __popc(mp & ltmask) + 1;  // inclusive cumsum
    const int rankN = negBase + offN + __popc(mn & ltmask) + 1;
    posBase += totP; negBase += totN;
    const bool psel = pos && (rankP <= np);
    const bool nsel = neg && (rankN <= nn);

    if (psel || nsel) {
      float x = obj[bN + i];
      float z = psel ? 1.0f : 0.0f;
      obj_acc += fmaxf(x, 0.0f) - x * z + log1pf(expf(-fabsf(x)));
    }
    if (psel) {
      float aw = a.z - a.x + 1.0f, ah = a.w - a.y + 1.0f;
      float axc = a.x + 0.5f * aw, ayc = a.y + 0.5f * ah;
      float gx1 = s_gt[4*arg+0], gy1 = s_gt[4*arg+1], gx2 = s_gt[4*arg+2], gy2 = s_gt[4*arg+3];
      float gw = gx2 - gx1 + 1.0f, gh = gy2 - gy1 + 1.0f;
      float gxc = gx1 + 0.5f * gw, gyc = gy1 + 0.5f * gh;
      float t0 = (gxc - axc) / aw, t1 = (gyc - ayc) / ah;
      float t2 = logf(gw / aw), t3 = logf(gh / ah);
      float4 r = breg[bN + i];
      float d;
      d = fabsf(r.x - t0); box_acc += (d < BETA_C) ? 0.5f * d * d / BETA_C : d - 0.5f * BETA_C;
      d = fabsf(r.y - t1); box_acc += (d < BETA_C) ? 0.5f * d * d / BETA_C : d - 0.5f * BETA_C;
      d = fabsf(r.z - t2); box_acc += (d < BETA_C) ? 0.5f * d * d / BETA_C : d - 0.5f * BETA_C;
      d = fabsf(r.w - t3); box_acc += (d < BETA_C) ? 0.5f * d * d / BETA_C : d - 0.5f * BETA_C;
    }
  }

  obj_acc = wave_sumf(obj_acc);
  box_acc = wave_sumf(box_acc);
  if (lane == 0) { s_ro[wv] = obj_acc; s_rb[wv] = box_acc; }
  __syncthreads();
  if (threadIdx.x == 0) {
    float so = 0.0f, sb = 0.0f;
    for (int w = 0; w < 8; ++w) { so += s_ro[w]; sb += s_rb[w]; }
    partials[(b * nblk + bx) * 2 + 0] = so;
    partials[(b * nblk + bx) * 2 + 1] = sb;
  }
}

__global__ void k_final(const int* __restrict__ numPN, const float* __restrict__ partials,
                        int nPart, float* __restrict__ out) {
  __shared__ float s_so[8], s_sb[8];
  float so = 0.0f, sb = 0.0f;
  for (int k = (int)threadIdx.x; k < nPart; k += BLOCK) {   // fixed strided order: deterministic
    so += partials[2*k];
    sb += partials[2*k+1];
  }
  so = wave_sumf(so);
  sb = wave_sumf(sb);
  const int lane = threadIdx.x & 31, wv = threadIdx.x >> 5;
  if (lane == 0) { s_so[wv] = so; s_sb[wv] = sb; }
  __syncthreads();
  if (threadIdx.x == 0) {
    float o = 0.0f, bx = 0.0f;
    for (int w = 0; w < 8; ++w) { o += s_so[w]; bx += s_sb[w]; }
    float tot = (float)(numPN[0] + numPN[1] + numPN[2] + numPN[3]);
    out[0] = o / tot;    // objectness_loss
    out[1] = bx / tot;   // box_loss
  }
}

// ---------------- host entry ----------------
extern "C" void kernel_launch(void* const* d_in, const int* in_sizes, int n_in,
                              void* d_out, int out_size, void* d_ws, size_t ws_size,
                              hipStream_t stream) {
  const float*  objectness = (const float*)d_in[0];   // [B,N]
  const float4* box_reg    = (const float4*)d_in[1];  // [B,N,4]
  const float4* anchors    = (const float4*)d_in[2];  // [B,N,4]
  const float*  gt_boxes   = (const float*)d_in[3];   // [B,M,4]
  const int*    visibility = (const int*)d_in[4];     // [B,N] (nonzero = visible)
  float* out = (float*)d_out;                         // [2]

  const int B = 2;
  const int N = in_sizes[0] / B;                      // 250000
  const int nblk = (N + CHUNK - 1) / CHUNK;           // 245 (<= 256 required by k_scan)

  // workspace partition (~13 KB total)
  unsigned* highest  = (unsigned*)d_ws;               // B*M floats (as uint bits)
  int* blockPos = (int*)((char*)d_ws + 512);
  int* blockNeg = blockPos + B * nblk;
  int* posOff   = blockNeg + B * nblk;
  int* negOff   = posOff   + B * nblk;
  int* numPN    = negOff   + B * nblk;                // [B][2]
  float* partials = (float*)(numPN + 8);              // [B*nblk][2]

  dim3 grid(nblk, B);
  k_init <<<dim3(1), dim3(128), 0, stream>>>(highest, B * M_GT);
  k_gtmax<<<grid, dim3(BLOCK), 0, stream>>>(anchors, gt_boxes, highest, N);
  k_count<<<grid, dim3(BLOCK), 0, stream>>>(anchors, gt_boxes, (const float*)highest,
                                            visibility, blockPos, blockNeg, N, nblk);
  k_scan <<<dim3(B), dim3(BLOCK), 0, stream>>>(blockPos, blockNeg, posOff, negOff, numPN, nblk);
  k_loss <<<grid, dim3(BLOCK), 0, stream>>>(objectness, box_reg, anchors, gt_boxes,
                                            (const float*)highest, visibility,
                                            posOff, negOff, numPN, partials, N, nblk);
  k_final<<<dim3(1), dim3(BLOCK), 0, stream>>>(numPN, partials, B * nblk, out);
}